// Lloss_16028817948844
// MI455X (gfx1250) — compile-verified
//
#include <hip/hip_runtime.h>
#include <hip/hip_bf16.h>

#define BATCH   8192
#define NSTOCK  4096
#define PER_LANE 16
#define SEG      (32 * PER_LANE)      // 512 elements per segment
#define NSEG     (NSTOCK / SEG)       // 8 segments per row
#define WAVES_PER_BLOCK 8

static __device__ __forceinline__ float ident_m() { return -1.0e30f; }

// Merge logsumexp states: (m,s) <- combine((m,s),(om,os)).
// Single exp via exp(-|d|) + selects; safe for the (-1e30, 0) identity.
static __device__ __forceinline__ void lse_combine(float& m, float& s, float om, float os) {
    float d  = om - m;
    float t  = __expf(-fabsf(d));
    bool  up = d > 0.0f;
    float s_up = fmaf(s,  t, os);   // new max is om
    float s_dn = fmaf(os, t, s);    // max unchanged
    s = up ? s_up : s_dn;
    m = up ? om : m;
}

__global__ __launch_bounds__(32 * WAVES_PER_BLOCK)
void row_suffix_lse_kernel(const float* __restrict__ f, float* __restrict__ row_l) {
    const int lane = threadIdx.x & 31;
    const int row  = blockIdx.x * WAVES_PER_BLOCK + (threadIdx.x >> 5);
    if (row >= BATCH) return;
    const float* __restrict__ rowp = f + (size_t)row * NSTOCK;

    // Carry state: logsumexp state of everything to the right of current segment.
    float m = ident_m(), s = 0.0f;
    float msum   = 0.0f;   // sum of per-position running max terms
    float logsum = 0.0f;   // sum of log(s_i), amortized via products of 8
    float fsum   = 0.0f;   // sum of f over this lane's positions

    // Preload the rightmost segment (software pipeline stage 0).
    float4 b0, b1, b2, b3;
    {
        const float4* p = (const float4*)(rowp + (NSEG - 1) * SEG + lane * PER_LANE);
        b0 = p[0]; b1 = p[1]; b2 = p[2]; b3 = p[3];
    }

    for (int seg = NSEG - 1; seg >= 0; --seg) {
        float x[PER_LANE];
        x[ 0]=b0.x; x[ 1]=b0.y; x[ 2]=b0.z; x[ 3]=b0.w;
        x[ 4]=b1.x; x[ 5]=b1.y; x[ 6]=b1.z; x[ 7]=b1.w;
        x[ 8]=b2.x; x[ 9]=b2.y; x[10]=b2.z; x[11]=b2.w;
        x[12]=b3.x; x[13]=b3.y; x[14]=b3.z; x[15]=b3.w;

        // gfx1250 prefetch (global_prefetch_b8) two segments ahead.
        if (seg > 1)
            __builtin_prefetch(rowp + (seg - 2) * SEG + lane * PER_LANE, 0, 0);

        // Preload next segment while we compute on this one.
        if (seg > 0) {
            const float4* p = (const float4*)(rowp + (seg - 1) * SEG + lane * PER_LANE);
            b0 = p[0]; b1 = p[1]; b2 = p[2]; b3 = p[3];
        }

        // (1) Per-lane chunk aggregate (order-independent).
        float cm = x[0];
#pragma unroll
        for (int j = 1; j < PER_LANE; ++j) cm = fmaxf(cm, x[j]);
        float cs = 0.0f;
#pragma unroll
        for (int j = 0; j < PER_LANE; ++j) cs += __expf(x[j] - cm);

        // (2) Inclusive suffix scan across the 32 lanes (wave32 shuffles).
        float im = cm, is = cs;
#pragma unroll
        for (int off = 1; off < 32; off <<= 1) {
            float om = __shfl_down(im, (unsigned)off, 32);
            float os = __shfl_down(is, (unsigned)off, 32);
            if (lane + off > 31) { om = ident_m(); os = 0.0f; }
            lse_combine(im, is, om, os);
        }
        // (3) Exclusive suffix state = inclusive state of lane+1.
        float em = __shfl_down(im, 1u, 32);
        float es = __shfl_down(is, 1u, 32);
        if (lane == 31) { em = ident_m(); es = 0.0f; }

        // Start state for this lane's in-chunk scan: right-neighbors + carry.
        float sm = em, ss = es;
        lse_combine(sm, ss, m, s);

        // (4) Reverse scan of the 16 owned elements; log amortized per 8.
        float prod = 1.0f;
#pragma unroll
        for (int j = PER_LANE - 1; j >= 8; --j) {
            float xx = x[j];
            float d  = xx - sm;
            float t  = __expf(-fabsf(d));
            bool  up = d > 0.0f;
            float s_up = fmaf(ss, t, 1.0f);
            float s_dn = ss + t;
            ss = up ? s_up : s_dn;
            sm = up ? xx : sm;
            msum += sm;
            prod *= ss;          // ss in [1, 4096]; prod of 8 <= 2^96, f32-safe
            fsum += xx;
        }
        logsum += __logf(prod);
        prod = 1.0f;
#pragma unroll
        for (int j = 7; j >= 0; --j) {
            float xx = x[j];
            float d  = xx - sm;
            float t  = __expf(-fabsf(d));
            bool  up = d > 0.0f;
            float s_up = fmaf(ss, t, 1.0f);
            float s_dn = ss + t;
            ss = up ? s_up : s_dn;
            sm = up ? xx : sm;
            msum += sm;
            prod *= ss;
            fsum += xx;
        }
        logsum += __logf(prod);

        // (5) New carry = lane 0's final state (full segment + old carry).
        m = __shfl(sm, 0, 32);
        s = __shfl(ss, 0, 32);
    }

    // Wave reduction of (sum of suffix_lse) and (sum of f).
    float lsum = msum + logsum;
#pragma unroll
    for (int off = 16; off > 0; off >>= 1) {
        lsum += __shfl_down(lsum, (unsigned)off, 32);
        fsum += __shfl_down(fsum, (unsigned)off, 32);
    }
    if (lane == 0) row_l[row] = lsum - fsum;
}

// Final deterministic mean over BATCH row losses. The 32 KiB of row losses is
// staged into LDS via the gfx1250 async copy engine (GLOBAL_LOAD_ASYNC_TO_LDS,
// tracked by ASYNCcnt) when the builtin is available; otherwise direct loads.
__global__ __launch_bounds__(256)
void reduce_mean_kernel(const float* __restrict__ row_l, float* __restrict__ out) {
    __shared__ float  stage[BATCH];     // 32 KiB of 320 KiB LDS
    __shared__ double sdata[256];
    const int t = threadIdx.x;

#if __has_builtin(__builtin_amdgcn_global_load_async_to_lds_b128)
    // Builtin signature (from clang diagnostic): pointers to 16-byte int
    // vectors in AS1 (global) / AS3 (LDS), plus imm offset and imm cpol.
    typedef int v4i_t __attribute__((vector_size(4 * sizeof(int))));
    typedef __attribute__((address_space(1))) v4i_t* gv4i_p;
    typedef __attribute__((address_space(3))) v4i_t* lv4i_p;
#pragma unroll
    for (int k = 0; k < BATCH / (256 * 4); ++k) {        // 8 async B128 per thread
        const int idx = (k * 256 + t) * 4;               // 16B per lane per op
        __builtin_amdgcn_global_load_async_to_lds_b128(
            (gv4i_p)(row_l + idx), (lv4i_p)(stage + idx), 0, 0);
    }
#if __has_builtin(__builtin_amdgcn_s_wait_asynccnt)
    __builtin_amdgcn_s_wait_asynccnt(0);
#else
    asm volatile("s_wait_asynccnt 0x0" ::: "memory");
#endif
    __syncthreads();                                     // LDS visible to all waves
#else
#pragma unroll
    for (int k = 0; k < BATCH / 256; ++k)
        stage[k * 256 + t] = row_l[k * 256 + t];
    __syncthreads();
#endif

    double acc = 0.0;
#pragma unroll 4
    for (int i = t; i < BATCH; i += 256) acc += (double)stage[i];
    sdata[t] = acc;
    __syncthreads();
#pragma unroll
    for (int st = 128; st > 0; st >>= 1) {
        if (t < st) sdata[t] += sdata[t + st];
        __syncthreads();
    }
    if (t == 0) out[0] = (float)(sdata[0] / (double)BATCH);
}

extern "C" void kernel_launch(void* const* d_in, const int* in_sizes, int n_in,
                              void* d_out, int out_size, void* d_ws, size_t ws_size,
                              hipStream_t stream) {
    const float* f = (const float*)d_in[0];           // [BATCH, NSTOCK] f32
    (void)in_sizes; (void)n_in; (void)out_size; (void)ws_size;
    float* row_l = (float*)d_ws;                      // BATCH floats = 32 KiB scratch

    row_suffix_lse_kernel<<<BATCH / WAVES_PER_BLOCK, 32 * WAVES_PER_BLOCK, 0, stream>>>(f, row_l);
    reduce_mean_kernel<<<1, 256, 0, stream>>>(row_l, (float*)d_out);
}